// KANLayer_15968688407293
// MI455X (gfx1250) — compile-verified
//
#include <hip/hip_runtime.h>
#include <hip/hip_bf16.h>

// ---------------------------------------------------------------------------
// KAN fused layer for MI455X (gfx1250, wave32).
//   out[b,o] = sum_i ( W2[i,o,:] . silu( W1[i,o,:,:] @ feats(x[b,i]) + B1[i,o,:] ) ) + sum_i B2[i,o]
// B=2048, Din=64, Dout=64, H=32, F=7 (padded to 8).
//
// GEMM1 runs on V_WMMA_F32_16X16X4_F32: M=16 batch rows, N=16 h-columns,
// K = feature dim split 0..3 / 4..7 (slot 7 zero-padded).
// Workspace layout (needs 4MB + 4MB + 256B):
//   feats  [B][Din][8]        fp32 (A-matrix source, K-padded)
//   w1pad  [Din][Dout][H][8]  fp32 (B-matrix source, K-padded)
//   b2sum  [Dout]             fp32 (sum_i B2[i,o])
// ---------------------------------------------------------------------------

typedef __attribute__((ext_vector_type(2))) float v2f;
typedef __attribute__((ext_vector_type(8))) float v8f;

#define DIN  64
#define DOUT 64
#define HH   32
#define FPAD 8

// silu(v) = v * sigmoid(v) = v * 0.5*(1 + tanh(v/2)) -> single TRANS op on gfx1250
__device__ __forceinline__ float fast_silu(float v) {
    float t;
    float xh = 0.5f * v;
#if __has_builtin(__builtin_amdgcn_tanhf)
    t = __builtin_amdgcn_tanhf(xh);
#else
    // gfx1250 v_tanh_f32; trailing v_nop satisfies the 1-op TRANS result hazard
    asm volatile("v_tanh_f32 %0, %1\n\tv_nop" : "=v"(t) : "v"(xh));
#endif
    return v * (0.5f * t + 0.5f);
}

// Prologue: build padded features, pad W1 to F=8, reduce B2 over i.
__global__ void kan_prep_kernel(const float* __restrict__ x,
                                const float* __restrict__ W1,
                                const float* __restrict__ B2,
                                float* __restrict__ feats,
                                float* __restrict__ w1pad,
                                float* __restrict__ b2sum,
                                int bdin) {
    int idx = blockIdx.x * blockDim.x + threadIdx.x;
    if (idx < bdin) {
        float xv = x[idx];
        float4* d = (float4*)(feats + (size_t)idx * FPAD);
        d[0] = make_float4(xv, __sinf(xv), __sinf(2.f * xv), __sinf(4.f * xv));
        d[1] = make_float4(__cosf(xv), __cosf(2.f * xv), __cosf(4.f * xv), 0.f);
    }
    if (idx < DIN * DOUT * HH) {
        const float* s = W1 + (size_t)idx * 7;
        float* d = w1pad + (size_t)idx * FPAD;
        d[0] = s[0]; d[1] = s[1]; d[2] = s[2]; d[3] = s[3];
        d[4] = s[4]; d[5] = s[5]; d[6] = s[6]; d[7] = 0.f;
    }
    if (idx < DOUT) {
        float s = 0.f;
        for (int i = 0; i < DIN; ++i) s += B2[i * DOUT + idx];
        b2sum[idx] = s;
    }
}

// Main fused kernel. Block = 256 threads = 8 waves.
// blockIdx.x : batch tile of 32 rows; blockIdx.y : group of 4 o-values.
// wave w: o_local = w>>1, h-half = w&1 (columns h0..h0+15).
__global__ __launch_bounds__(256) void kan_fused_kernel(
    const float* __restrict__ feats,   // [B][Din][8]
    const float* __restrict__ w1pad,   // [Din][Dout][H][8]
    const float* __restrict__ W2,      // [Din][Dout][H]
    const float* __restrict__ B1,      // [Din][Dout][H]
    const float* __restrict__ b2sum,   // [Dout]
    float* __restrict__ out)           // [B][Dout]
{
    __shared__ float smem[32 * DIN * FPAD];   // 64KB feats tile; reused for output combine

    const int tid = threadIdx.x;
    const int b0  = blockIdx.x * 32;

    // Stage feats rows b0..b0+31 (all i, all 8 features) into LDS: 16384 floats.
    {
        const float4* src = (const float4*)(feats + (size_t)b0 * DIN * FPAD);
        float4* dst = (float4*)smem;
        #pragma unroll
        for (int k = 0; k < 16; ++k) dst[tid + k * 256] = src[tid + k * 256];
    }
    __syncthreads();

    const int w       = tid >> 5;
    const int lane    = tid & 31;
    const int lane16  = lane & 15;
    const int hi      = lane >> 4;        // K-group of this lane per ISA f32 A/B layout
    const int koff    = hi * 2;           // lane holds K = koff + vgpr
    const int o_local = w >> 1;
    const int o       = blockIdx.y * 4 + o_local;
    const int h0      = (w & 1) * 16;
    const int col     = h0 + lane16;      // h column owned by this lane (B/C N index)

    float accA[8], accB[8];
    #pragma unroll
    for (int g = 0; g < 8; ++g) { accA[g] = 0.f; accB[g] = 0.f; }

    const float* wbase = w1pad + ((size_t)o * HH + col) * FPAD;
    const float* sA0 = smem + (size_t)(lane16       * DIN) * FPAD + koff;
    const float* sA1 = smem + (size_t)((16 + lane16) * DIN) * FPAD + koff;

    for (int i = 0; i < DIN; ++i) {
        const float* wb = wbase + (size_t)i * (DOUT * HH * FPAD);
        v2f bk0 = *(const v2f*)(wb + koff);       // B: K = koff..koff+1
        v2f bk1 = *(const v2f*)(wb + koff + 4);   // B: K = 4+koff..
        v2f aA0 = *(const v2f*)(sA0 + i * FPAD);
        v2f aA1 = *(const v2f*)(sA0 + i * FPAD + 4);
        v2f aB0 = *(const v2f*)(sA1 + i * FPAD);
        v2f aB1 = *(const v2f*)(sA1 + i * FPAD + 4);

        if (i + 1 < DIN) __builtin_prefetch(wb + DOUT * HH * FPAD, 0, 0);

        v8f c0 = {}; v8f c1 = {};
        c0 = __builtin_amdgcn_wmma_f32_16x16x4_f32(false, aA0, false, bk0, (short)0, c0, false, false);
        c0 = __builtin_amdgcn_wmma_f32_16x16x4_f32(false, aA1, false, bk1, (short)0, c0, false, false);
        c1 = __builtin_amdgcn_wmma_f32_16x16x4_f32(false, aB0, false, bk0, (short)0, c1, false, false);
        c1 = __builtin_amdgcn_wmma_f32_16x16x4_f32(false, aB1, false, bk1, (short)0, c1, false, false);

        const int sidx = (i * DOUT + o) * HH + col;
        const float b1v = B1[sidx];
        const float w2v = W2[sidx];
        #pragma unroll
        for (int g = 0; g < 8; ++g) {
            accA[g] += fast_silu(c0[g] + b1v) * w2v;
            accB[g] += fast_silu(c1[g] + b1v) * w2v;
        }
    }

    // Reduce over the 16 h-columns held by this wave's lanes (per half).
    #pragma unroll
    for (int mask = 1; mask < 16; mask <<= 1) {
        #pragma unroll
        for (int g = 0; g < 8; ++g) {
            accA[g] += __shfl_xor(accA[g], mask, 32);
            accB[g] += __shfl_xor(accB[g], mask, 32);
        }
    }

    __syncthreads();                 // feats tile no longer needed; reuse smem
    if (tid < 128) smem[tid] = 0.f;  // out tile: [4 o_local][32 rows]
    __syncthreads();

    if (lane16 == 0) {
        // lane 0: C rows g (accA) / 16+g (accB); lane 16: rows 8+g / 24+g
        const int rbase = o_local * 32 + hi * 8;
        #pragma unroll
        for (int g = 0; g < 8; ++g) {
            atomicAdd(&smem[rbase + g],      accA[g]);
            atomicAdd(&smem[rbase + 16 + g], accB[g]);
        }
    }
    __syncthreads();

    if (tid < 128) {
        const int r  = tid & 31;
        const int ol = tid >> 5;
        const int oo = blockIdx.y * 4 + ol;
        out[(size_t)(b0 + r) * DOUT + oo] = smem[tid] + b2sum[oo];
    }
}

extern "C" void kernel_launch(void* const* d_in, const int* in_sizes, int n_in,
                              void* d_out, int out_size, void* d_ws, size_t ws_size,
                              hipStream_t stream) {
    const float* x  = (const float*)d_in[0];   // [B, Din]
    const float* W1 = (const float*)d_in[1];   // [Din, Dout, H, 7]
    const float* W2 = (const float*)d_in[2];   // [Din, Dout, 1, H]
    const float* B1 = (const float*)d_in[3];   // [Din, Dout, H]
    const float* B2 = (const float*)d_in[4];   // [Din, Dout, 1]
    float* out = (float*)d_out;                // [B, Dout]

    const int B    = in_sizes[0] / DIN;        // 2048
    const int bdin = B * DIN;

    float* feats = (float*)d_ws;                              // B*Din*8 floats (4MB)
    float* w1pad = feats + (size_t)bdin * FPAD;               // Din*Dout*H*8 floats (4MB)
    float* b2sum = w1pad + (size_t)DIN * DOUT * HH * FPAD;    // Dout floats

    const int prepN  = (bdin > DIN * DOUT * HH) ? bdin : DIN * DOUT * HH;
    const int prepBl = (prepN + 255) / 256;
    kan_prep_kernel<<<prepBl, 256, 0, stream>>>(x, W1, B2, feats, w1pad, b2sum, bdin);

    dim3 grid(B / 32, DOUT / 4);
    kan_fused_kernel<<<grid, 256, 0, stream>>>(feats, w1pad, W2, B1, b2sum, out);
}